// GCN_single_output_regression_70171175682690
// MI455X (gfx1250) — compile-verified
//
#include <hip/hip_runtime.h>

typedef __attribute__((ext_vector_type(2))) float v2f;
typedef __attribute__((ext_vector_type(8))) float v8f;

#define TPB 256

// ---------------------------------------------------------------------------
// K0: deg = 1 (self loop), acc1 = 0, acc2 = 0   (d_ws is poisoned, must init)
// ---------------------------------------------------------------------------
__global__ void k_init(float* __restrict__ deg, float* __restrict__ acc1,
                       float* __restrict__ acc2, int n) {
  int i = blockIdx.x * blockDim.x + threadIdx.x;
  if (i < 16 * n) acc1[i] = 0.0f;
  if (i < n) { deg[i] = 1.0f; acc2[i] = 0.0f; }
}

// ---------------------------------------------------------------------------
// K1: deg[col[e]] += 1   (4M fire-and-forget f32 atomics, L2-resident)
// ---------------------------------------------------------------------------
__global__ void k_degree(const int* __restrict__ col, float* __restrict__ deg, int ne) {
  int e = blockIdx.x * blockDim.x + threadIdx.x;
  if (e < ne) atomicAdd(&deg[col[e]], 1.0f);
}

// ---------------------------------------------------------------------------
// K2: deg -> rsqrt(deg) in place (deg >= 1 guaranteed by self loops)
// ---------------------------------------------------------------------------
__global__ void k_dinv(float* __restrict__ deg, int n) {
  int i = blockIdx.x * blockDim.x + threadIdx.x;
  if (i < n) deg[i] = rsqrtf(deg[i]);
}

// ---------------------------------------------------------------------------
// K3: hs1[m][j] = dinv[m] * sum_k x[m][k] * W1[k][j]   via V_WMMA_F32_16X16X4_F32
//     One wave32 per 16-node tile: A = 16x4 (K=3 padded), B = 4x16 = W1,
//     C/D = 16x16 f32 accumulator (8 VGPRs).
//     A layout:  lanes 0-15 hold K=0,1 in v[0],v[1]; lanes 16-31 hold K=2,3.
//     B layout:  lanes 0-15 hold rows K=0,1 (N=lane); lanes 16-31 rows K=2,3.
//     D layout:  lane<16 -> (M=r, N=lane); lane>=16 -> (M=r+8, N=lane-16).
//     Operand setup is select-based (no EXEC divergence); store loop has a
//     wave-uniform full-tile fast path so the hot path is branchless.
// ---------------------------------------------------------------------------
__global__ void k_h1_wmma(const float* __restrict__ x, const float* __restrict__ W1,
                          const float* __restrict__ dinv, float* __restrict__ hs1, int n) {
  int wave = (blockIdx.x * blockDim.x + threadIdx.x) >> 5;
  int lane = threadIdx.x & 31;
  int base = wave * 16;
  if (base >= n) return;              // wave-uniform: EXEC stays all-ones for WMMA

  int nn = lane & 15;                 // column / node-in-tile selector
  int hi = lane >> 4;                 // 0: lanes 0-15, 1: lanes 16-31
  int m  = base + nn;
  int mr = (m < n) ? m : (n - 1);     // clamp reads in tail tile (select, no branch)

  int k0 = hi ? 2 : 0;                // first K handled by this lane half
  float pad = hi ? 0.0f : 1.0f;       // K=3 column is zero padding

  // A matrix (nodes x features_in): unconditional loads + select for padding
  float ax = x[mr * 3 + k0];          // K=0 (lo) or K=2 (hi)
  float ay = x[mr * 3 + 1] * pad;     // K=1 (lo) or 0   (hi)
  v2f a; a.x = ax; a.y = ay;

  // B matrix = W1 (4 x 16, row K=3 is zero padding)
  float bx = W1[k0 * 16 + nn];        // row K=0 (lo) or K=2 (hi)
  float by = W1[1 * 16 + nn] * pad;   // row K=1 (lo) or 0  (hi)
  v2f b; b.x = bx; b.y = by;

  v8f c = {};
  c = __builtin_amdgcn_wmma_f32_16x16x4_f32(
      /*neg_a=*/false, a, /*neg_b=*/false, b,
      /*c_mod=*/(short)0, c, /*reuse_a=*/false, /*reuse_b=*/false);

  // Scale each row by dinv[node] and store row-major hs1[node][16]
  int rbase = base + (hi << 3);
  if (base + 16 <= n) {               // wave-uniform fast path: no guards
#pragma unroll
    for (int r = 0; r < 8; ++r) {
      int node = rbase + r;
      hs1[node * 16 + nn] = c[r] * dinv[node];
    }
  } else {                            // tail tile only
#pragma unroll
    for (int r = 0; r < 8; ++r) {
      int node = rbase + r;
      if (node < n) hs1[node * 16 + nn] = c[r] * dinv[node];
    }
  }
}

// ---------------------------------------------------------------------------
// K4: layer-1 scatter: acc1[col][j] += hs1[row][j]
//     16 threads per edge -> 64B-coalesced gather + 64B-coalesced atomics.
//     Index loads are 16x-redundant but broadcast-cached in WGP$/L2.
// ---------------------------------------------------------------------------
__global__ void k_edge1(const int* __restrict__ row, const int* __restrict__ col,
                        const float* __restrict__ hs1, float* __restrict__ acc1, int ne) {
  int gid = blockIdx.x * blockDim.x + threadIdx.x;
  int e = gid >> 4;
  int j = gid & 15;
  if (e >= ne) return;
  int r = row[e];
  int c = col[e];
  atomicAdd(&acc1[c * 16 + j], hs1[r * 16 + j]);
}

// ---------------------------------------------------------------------------
// K5: per node: out1 = relu(b1 + dinv*(acc1 + hs1));  gs = (out1 . W2) * dinv
//     Fuses layer-1 epilogue + ReLU + layer-2 linear + layer-2 pre-scale,
//     so the N x 16 hidden activation is consumed in-register.
// ---------------------------------------------------------------------------
__global__ void k_fuse(const float4* __restrict__ acc1, const float4* __restrict__ hs1,
                       const float* __restrict__ dinv, const float* __restrict__ b1,
                       const float* __restrict__ W2, float* __restrict__ gs, int n) {
  int i = blockIdx.x * blockDim.x + threadIdx.x;
  if (i >= n) return;
  float di = dinv[i];
  float g = 0.0f;
#pragma unroll
  for (int q = 0; q < 4; ++q) {
    float4 a = acc1[i * 4 + q];
    float4 h = hs1[i * 4 + q];
    float o;
    o = fmaxf(fmaf(di, a.x + h.x, b1[4 * q + 0]), 0.0f); g = fmaf(o, W2[4 * q + 0], g);
    o = fmaxf(fmaf(di, a.y + h.y, b1[4 * q + 1]), 0.0f); g = fmaf(o, W2[4 * q + 1], g);
    o = fmaxf(fmaf(di, a.z + h.z, b1[4 * q + 2]), 0.0f); g = fmaf(o, W2[4 * q + 2], g);
    o = fmaxf(fmaf(di, a.w + h.w, b1[4 * q + 3]), 0.0f); g = fmaf(o, W2[4 * q + 3], g);
  }
  gs[i] = g * di;
}

// ---------------------------------------------------------------------------
// K6: layer-2 scatter: acc2[col] += gs[row]
// ---------------------------------------------------------------------------
__global__ void k_edge2(const int* __restrict__ row, const int* __restrict__ col,
                        const float* __restrict__ gs, float* __restrict__ acc2, int ne) {
  int e = blockIdx.x * blockDim.x + threadIdx.x;
  if (e >= ne) return;
  atomicAdd(&acc2[col[e]], gs[row[e]]);
}

// ---------------------------------------------------------------------------
// K7: out[i] = b2 + dinv[i] * (acc2[i] + gs[i])   (self-loop folded in)
// ---------------------------------------------------------------------------
__global__ void k_out(const float* __restrict__ acc2, const float* __restrict__ gs,
                      const float* __restrict__ dinv, const float* __restrict__ b2,
                      float* __restrict__ out, int n) {
  int i = blockIdx.x * blockDim.x + threadIdx.x;
  if (i < n) out[i] = fmaf(dinv[i], acc2[i] + gs[i], b2[0]);
}

// ---------------------------------------------------------------------------
extern "C" void kernel_launch(void* const* d_in, const int* in_sizes, int n_in,
                              void* d_out, int out_size, void* d_ws, size_t ws_size,
                              hipStream_t stream) {
  const float* x  = (const float*)d_in[0];
  const int*   ei = (const int*)d_in[1];
  const float* W1 = (const float*)d_in[2];
  const float* b1 = (const float*)d_in[3];
  const float* W2 = (const float*)d_in[4];
  const float* b2 = (const float*)d_in[5];

  const int n = in_sizes[0] / 3;       // 500,000 nodes
  const int E = in_sizes[1] / 2;       // 4,000,000 edges
  const int* row = ei;                 // edge_index[0] = sources
  const int* col = ei + E;             // edge_index[1] = targets

  // Workspace layout (16B-aligned stripes), ~35*N floats = ~70 MB:
  size_t Np = ((size_t)n + 3) & ~(size_t)3;
  float* ws   = (float*)d_ws;
  float* deg  = ws;                    // [n]    degree -> dinv (in place)
  float* hs1  = ws + Np;               // [16n]  dinv-scaled x@W1
  float* acc1 = ws + 17 * Np;          // [16n]  layer-1 scatter accumulator
  float* gs   = ws + 33 * Np;          // [n]    dinv-scaled layer-2 messages
  float* acc2 = ws + 34 * Np;          // [n]    layer-2 scatter accumulator

  const int tpb = TPB;
  const unsigned gN   = (unsigned)((n + tpb - 1) / tpb);
  const unsigned g16N = (unsigned)(((size_t)16 * n + tpb - 1) / tpb);
  const unsigned gE   = (unsigned)((E + tpb - 1) / tpb);
  const unsigned g16E = (unsigned)(((size_t)16 * E + tpb - 1) / tpb);
  const int waves_per_blk = tpb / 32;
  const unsigned gH1 = (unsigned)(((n + 15) / 16 + waves_per_blk - 1) / waves_per_blk);

  k_init  <<<g16N, tpb, 0, stream>>>(deg, acc1, acc2, n);
  k_degree<<<gE,   tpb, 0, stream>>>(col, deg, E);
  k_dinv  <<<gN,   tpb, 0, stream>>>(deg, n);
  k_h1_wmma<<<gH1, tpb, 0, stream>>>(x, W1, deg, hs1, n);
  k_edge1 <<<g16E, tpb, 0, stream>>>(row, col, hs1, acc1, E);
  k_fuse  <<<gN,   tpb, 0, stream>>>((const float4*)acc1, (const float4*)hs1,
                                     deg, b1, W2, gs, n);
  k_edge2 <<<gE,   tpb, 0, stream>>>(row, col, gs, acc2, E);
  k_out   <<<gN,   tpb, 0, stream>>>(acc2, gs, deg, b2, (float*)d_out, n);
}